// SmallGCN_5557687681608
// MI455X (gfx1250) — compile-verified
//
#include <hip/hip_runtime.h>
#include <hip/hip_bf16.h>

typedef float v2f __attribute__((ext_vector_type(2)));
typedef float v8f __attribute__((ext_vector_type(8)));

#define FEAT1 128
#define FEAT2 40

// Non-returning device-scope float atomic add, guaranteed native
// (GLOBAL_ATOMIC_ADD_F32, TH=0 non-returning, SCOPE=DEV for cross-WG coherence).
static __device__ __forceinline__ void atomic_add_f32(float* p, float v) {
  asm volatile("global_atomic_add_f32 %0, %1, off scope:SCOPE_DEV"
               :
               : "v"((unsigned long long)(uintptr_t)p), "v"(v)
               : "memory");
}

// ---------------- degree / norm ----------------
__global__ void k_init_deg(float* deg, int n) {
  int i = blockIdx.x * blockDim.x + threadIdx.x;
  if (i < n) deg[i] = 1.0f;  // self-loop contribution
}

__global__ void k_degree(const int* __restrict__ col, float* deg, int E) {
  int e = blockIdx.x * blockDim.x + threadIdx.x;
  if (e < E) atomic_add_f32(&deg[col[e]], 1.0f);
}

__global__ void k_dinv(float* deg, int n) {
  int i = blockIdx.x * blockDim.x + threadIdx.x;
  if (i < n) {
    float d = deg[i];
    deg[i] = (d > 0.0f) ? rsqrtf(d) : 0.0f;
  }
}

// ---------------- init buffers ----------------
__global__ void k_zero4(float4* p, long n4) {
  long i = blockIdx.x * (long)blockDim.x + threadIdx.x;
  if (i < n4) p[i] = make_float4(0.f, 0.f, 0.f, 0.f);
}

__global__ void k_init_out(float* out, const float* __restrict__ b2, long total) {
  long i = blockIdx.x * (long)blockDim.x + threadIdx.x;
  if (i < total) out[i] = b2[i % FEAT2];
}

// ---------------- GEMM1: h1 = x @ W1  (N x 128) ----------------
// One wave computes one 16x16 f32 tile via V_WMMA_F32_16X16X4_F32, K stepped by 4.
// Layouts per ISA 7.12.2: A 16x4 (lanes 0-15: M, VGPR0/1 = K0/K1; lanes 16-31:
// K2/K3), B mirrored, C/D rows striped over 8 VGPRs with M split 0-7 / 8-15.
__global__ void k_gemm1(const float* __restrict__ x, const float* __restrict__ W,
                        float* __restrict__ h, int mtiles) {
  const int lane = threadIdx.x & 31;
  const int wid  = threadIdx.x >> 5;
  int t = blockIdx.x * (blockDim.x >> 5) + wid;
  if (t >= mtiles * 8) return;                 // wave-uniform: EXEC stays full
  const int mt = t >> 3, nt = t & 7;
  const int m0 = mt << 4, n0 = nt << 4;
  const int lm = lane & 15;                    // M index (A) / N index (B,C,D)
  const int hi = lane >> 4;                    // K half select

  const float* arow = x + (size_t)(m0 + lm) * FEAT1 + 2 * hi;
  const float* bcol = W + n0 + lm;

  v8f acc = {0.f, 0.f, 0.f, 0.f, 0.f, 0.f, 0.f, 0.f};
#pragma unroll 8
  for (int k = 0; k < FEAT1; k += 4) {
    v2f a = *(const v2f*)(arow + k);           // A[m][k+2hi], A[m][k+2hi+1]
    const int kb = k + 2 * hi;
    v2f b;
    b.x = bcol[(size_t)kb * FEAT1];            // B[kb][n]
    b.y = bcol[(size_t)(kb + 1) * FEAT1];      // B[kb+1][n]
    acc = __builtin_amdgcn_wmma_f32_16x16x4_f32(false, a, false, b,
                                                (short)0, acc, false, false);
  }
  const int mbase = m0 + hi * 8;
  float* dst = h + (size_t)mbase * FEAT1 + n0 + lm;
#pragma unroll
  for (int r = 0; r < 8; ++r) dst[(size_t)r * FEAT1] = acc[r];
}

// ---------------- GEMM2: h2 = relu(agg1 + b1) @ W2  (N x 40, padded to 48) ----
__global__ void k_gemm2(const float* __restrict__ agg, const float* __restrict__ b1,
                        const float* __restrict__ W2, float* __restrict__ h2,
                        int mtiles) {
  const int lane = threadIdx.x & 31;
  const int wid  = threadIdx.x >> 5;
  int t = blockIdx.x * (blockDim.x >> 5) + wid;
  if (t >= mtiles * 3) return;                 // wave-uniform
  const int mt = t / 3, nt = t % 3;
  const int m0 = mt << 4, n0 = nt << 4;
  const int lm = lane & 15;
  const int hi = lane >> 4;
  const int n  = n0 + lm;

  const float* arow = agg + (size_t)(m0 + lm) * FEAT1 + 2 * hi;

  v8f acc = {0.f, 0.f, 0.f, 0.f, 0.f, 0.f, 0.f, 0.f};
  for (int k = 0; k < FEAT1; k += 4) {
    const int ka = k + 2 * hi;
    v2f a;
    a.x = fmaxf(arow[k]     + b1[ka],     0.0f);   // fused bias + ReLU
    a.y = fmaxf(arow[k + 1] + b1[ka + 1], 0.0f);
    v2f b;
    if (n < FEAT2) {
      b.x = W2[(size_t)ka * FEAT2 + n];
      b.y = W2[(size_t)(ka + 1) * FEAT2 + n];
    } else {
      b.x = 0.f; b.y = 0.f;                        // zero-pad columns 40..47
    }
    acc = __builtin_amdgcn_wmma_f32_16x16x4_f32(false, a, false, b,
                                                (short)0, acc, false, false);
  }
  if (n < FEAT2) {
    const int mbase = m0 + hi * 8;
    float* dst = h2 + (size_t)mbase * FEAT2 + n;
#pragma unroll
    for (int r = 0; r < 8; ++r) dst[(size_t)r * FEAT2] = acc[r];
  }
}

// ---------------- scatter layer 1: agg1[col] += h1[row]*norm (128 feats) -----
// One wave per edge; lane handles 4 consecutive features (float4 gather).
__global__ void k_scatter1(const int* __restrict__ row, const int* __restrict__ col,
                           const float* __restrict__ dinv, const float* __restrict__ h1,
                           float* agg, int E, int N) {
  const int gw   = blockIdx.x * (blockDim.x >> 5) + (threadIdx.x >> 5);
  const int lane = threadIdx.x & 31;
  if (gw >= E + N) return;
  int r, c;
  if (gw < E) { r = row[gw]; c = col[gw]; }
  else        { r = gw - E;  c = r; }            // self-loop edges
  const float w = dinv[r] * dinv[c];
  const float4 v = ((const float4*)(h1 + (size_t)r * FEAT1))[lane];
  float* dst = agg + (size_t)c * FEAT1 + lane * 4;
  atomic_add_f32(dst + 0, v.x * w);
  atomic_add_f32(dst + 1, v.y * w);
  atomic_add_f32(dst + 2, v.z * w);
  atomic_add_f32(dst + 3, v.w * w);
}

// ---------------- scatter layer 2: out[col] += h2[row]*norm (40 feats) -------
__global__ void k_scatter2(const int* __restrict__ row, const int* __restrict__ col,
                           const float* __restrict__ dinv, const float* __restrict__ h2,
                           float* out, int E, int N) {
  const int gw   = blockIdx.x * (blockDim.x >> 5) + (threadIdx.x >> 5);
  const int lane = threadIdx.x & 31;
  if (gw >= E + N) return;
  int r, c;
  if (gw < E) { r = row[gw]; c = col[gw]; }
  else        { r = gw - E;  c = r; }
  const float w = dinv[r] * dinv[c];
  const float* src = h2 + (size_t)r * FEAT2;
  float* dst = out + (size_t)c * FEAT2;
  for (int f = lane; f < FEAT2; f += 32)
    atomic_add_f32(dst + f, src[f] * w);
}

extern "C" void kernel_launch(void* const* d_in, const int* in_sizes, int n_in,
                              void* d_out, int out_size, void* d_ws, size_t ws_size,
                              hipStream_t stream) {
  const float* x  = (const float*)d_in[0];
  const int*   ei = (const int*)d_in[1];
  const float* W1 = (const float*)d_in[2];
  const float* b1 = (const float*)d_in[3];
  const float* W2 = (const float*)d_in[4];
  const float* b2 = (const float*)d_in[5];
  float* out = (float*)d_out;

  const int N = in_sizes[0] / FEAT1;   // 50000
  const int E = in_sizes[1] / 2;       // 800000
  const int* row = ei;
  const int* col = ei + E;

  float* ws  = (float*)d_ws;
  float* deg = ws;                               // N   (becomes dinv in place)
  float* h1  = deg + N;                          // N*128
  float* agg = h1 + (size_t)N * FEAT1;           // N*128
  float* h2  = agg + (size_t)N * FEAT1;          // N*40

  const int TB = 256;

  // degrees -> dinv (deg initialized to 1.0 for the self-loop)
  k_init_deg<<<(N + TB - 1) / TB, TB, 0, stream>>>(deg, N);
  k_degree<<<(E + TB - 1) / TB, TB, 0, stream>>>(col, deg, E);
  k_dinv<<<(N + TB - 1) / TB, TB, 0, stream>>>(deg, N);

  // zero agg1; seed out with b2 (fuses the +b2 into the scatter accumulation)
  long n4 = (long)N * FEAT1 / 4;
  k_zero4<<<(int)((n4 + TB - 1) / TB), TB, 0, stream>>>((float4*)agg, n4);
  long nout = (long)N * FEAT2;
  k_init_out<<<(int)((nout + TB - 1) / TB), TB, 0, stream>>>(out, b2, nout);

  // GEMM1 (WMMA f32): 3125 x 8 tiles, 4 waves per 128-thread block
  const int mtiles = N / 16;
  const int t1 = mtiles * 8;
  k_gemm1<<<(t1 + 3) / 4, 128, 0, stream>>>(x, W1, h1, mtiles);

  // scatter layer 1 (one wave per edge incl. self-loops)
  const long waves = (long)E + N;
  const int wpb = TB / 32;
  k_scatter1<<<(int)((waves + wpb - 1) / wpb), TB, 0, stream>>>(row, col, deg, h1, agg, E, N);

  // GEMM2 (WMMA f32, fused bias+ReLU on A): 3125 x 3 tiles
  const int t2 = mtiles * 3;
  k_gemm2<<<(t2 + 3) / 4, 128, 0, stream>>>(agg, b1, W2, h2, mtiles);

  // scatter layer 2 into out (pre-seeded with b2)
  k_scatter2<<<(int)((waves + wpb - 1) / wpb), TB, 0, stream>>>(row, col, deg, h2, out, E, N);
}